// MSE_Var_Loss_32195074851231
// MI455X (gfx1250) — compile-verified
//
#include <hip/hip_runtime.h>
#include <stdint.h>

typedef __attribute__((ext_vector_type(2))) float v2f;
typedef __attribute__((ext_vector_type(8))) float v8f;

#define TPB 256

// ---- Pade-13 coefficients (float) ----
#define B0  6.476475253248e+16f
#define B1  3.238237626624e+16f
#define B2  7.7717703038976e+15f
#define B3  1.1873537964288e+15f
#define B4  1.29060195264e+14f
#define B5  1.05594705216e+13f
#define B6  6.704425728e+11f
#define B7  3.352212864e+10f
#define B8  1.32324192e+9f
#define B9  4.08408e+7f
#define B10 960960.0f
#define B11 16380.0f
#define B12 182.0f
#define B13 1.0f
#define INV_MAXNORM 0.18615310290518018f  // 1 / 5.371920351148152

__device__ __forceinline__ void mm3(const float* __restrict__ A,
                                    const float* __restrict__ B,
                                    float* __restrict__ C) {
#pragma unroll
  for (int r = 0; r < 3; ++r) {
#pragma unroll
    for (int c = 0; c < 3; ++c) {
      C[3 * r + c] = fmaf(A[3 * r + 0], B[0 + c],
                     fmaf(A[3 * r + 1], B[3 + c],
                          A[3 * r + 2] * B[6 + c]));
    }
  }
}

__device__ __forceinline__ uint32_t lds_off(const void* p) {
  // generic LDS pointer: low 32 bits are the wave-relative LDS byte offset
  return (uint32_t)(uintptr_t)p;
}

__device__ __forceinline__ void async_b128(uint32_t lds_byte, uint32_t goff_byte,
                                           const float* sbase) {
  asm volatile("global_load_async_to_lds_b128 %0, %1, %2"
               :: "v"(lds_byte), "v"(goff_byte), "s"(sbase)
               : "memory");
}

__global__ __launch_bounds__(TPB) void expm_loss_kernel(
    const float* __restrict__ mean, const float* __restrict__ var,
    const float* __restrict__ gt, float* __restrict__ blockSums,
    long long total) {
  __shared__ float shVar[TPB * 9];
  __shared__ float shMean[TPB * 3];
  __shared__ float shGt[TPB * 3];
  __shared__ float warpSums[TPB / 32];

  const int t = threadIdx.x;
  const long long blockStart = (long long)blockIdx.x * TPB;

  // ---- async stage: coalesced 128-bit global -> LDS (ASYNCcnt path) ----
  {
    const float* base = var + blockStart * 9;
    const long long remV = total * 9 - blockStart * 9;
    uint32_t lb = lds_off(&shVar[0]);
#pragma unroll
    for (int i = t; i < TPB * 9 / 4; i += TPB) {  // 576 float4s
      if ((long long)(i * 4 + 4) <= remV) {
        uint32_t off = (uint32_t)(i * 16);
        async_b128(lb + off, off, base);
      }
    }
  }
  {
    const float* base = mean + blockStart * 3;
    const long long remM = total * 3 - blockStart * 3;
    uint32_t lb = lds_off(&shMean[0]);
    if (t < TPB * 3 / 4) {  // 192 float4s
      if ((long long)(t * 4 + 4) <= remM) {
        uint32_t off = (uint32_t)(t * 16);
        async_b128(lb + off, off, base);
      }
    }
  }
  {
    const float* base = gt + blockStart * 3;
    const long long remG = total * 3 - blockStart * 3;
    uint32_t lb = lds_off(&shGt[0]);
    if (t < TPB * 3 / 4) {
      if ((long long)(t * 4 + 4) <= remG) {
        uint32_t off = (uint32_t)(t * 16);
        async_b128(lb + off, off, base);
      }
    }
  }
  asm volatile("s_wait_asynccnt 0x0" ::: "memory");
  __syncthreads();

  const long long idx = blockStart + t;
  const bool ok = idx < total;

  // ---- per-thread registers (bank-conflict-free LDS reads: strides 9,3 coprime to 64) ----
  float m[9];
#pragma unroll
  for (int i = 0; i < 9; ++i) m[i] = shVar[t * 9 + i];
  float dlt[3];
#pragma unroll
  for (int i = 0; i < 3; ++i) dlt[i] = shGt[t * 3 + i] - shMean[t * 3 + i];

  // ---- scaling: n_sq = max(0, ceil(log2(||A||_F / maxnorm))) ----
  float ss = 0.f;
#pragma unroll
  for (int i = 0; i < 9; ++i) ss = fmaf(m[i], m[i], ss);
  const float fro = sqrtf(ss);
  const float nsq = fmaxf(0.0f, ceilf(log2f(fro * INV_MAXNORM)));
  const float sc = exp2f(-nsq);
  const int nsi = (int)nsq;

  float As[9];
#pragma unroll
  for (int i = 0; i < 9; ++i) As[i] = m[i] * sc;

  // ---- Pade-13 numerator/denominator ----
  float A2[9], A4[9], A6[9];
  mm3(As, As, A2);
  mm3(A2, A2, A4);
  mm3(A4, A2, A6);

  float W[9], T[9], U[9], V[9];
#pragma unroll
  for (int i = 0; i < 9; ++i) W[i] = fmaf(B13, A6[i], fmaf(B11, A4[i], B9 * A2[i]));
  mm3(A6, W, T);
#pragma unroll
  for (int i = 0; i < 9; ++i) T[i] += fmaf(B7, A6[i], fmaf(B5, A4[i], B3 * A2[i]));
  T[0] += B1; T[4] += B1; T[8] += B1;
  mm3(As, T, U);

#pragma unroll
  for (int i = 0; i < 9; ++i) W[i] = fmaf(B12, A6[i], fmaf(B10, A4[i], B8 * A2[i]));
  mm3(A6, W, V);
#pragma unroll
  for (int i = 0; i < 9; ++i) V[i] += fmaf(B6, A6[i], fmaf(B4, A4[i], B2 * A2[i]));
  V[0] += B0; V[4] += B0; V[8] += B0;

  float P[9], Q[9];
#pragma unroll
  for (int i = 0; i < 9; ++i) { P[i] = U[i] + V[i]; Q[i] = V[i] - U[i]; }

  // ---- R = Q^{-1} P  (3x3 Cramer / adjugate) ----
  const float c00 = Q[4] * Q[8] - Q[5] * Q[7];
  const float c01 = Q[5] * Q[6] - Q[3] * Q[8];
  const float c02 = Q[3] * Q[7] - Q[4] * Q[6];
  const float detQ = fmaf(Q[0], c00, fmaf(Q[1], c01, Q[2] * c02));
  const float invd = 1.0f / detQ;
  float iQ[9];
  iQ[0] = c00 * invd;
  iQ[1] = (Q[2] * Q[7] - Q[1] * Q[8]) * invd;
  iQ[2] = (Q[1] * Q[5] - Q[2] * Q[4]) * invd;
  iQ[3] = c01 * invd;
  iQ[4] = (Q[0] * Q[8] - Q[2] * Q[6]) * invd;
  iQ[5] = (Q[2] * Q[3] - Q[0] * Q[5]) * invd;
  iQ[6] = c02 * invd;
  iQ[7] = (Q[1] * Q[6] - Q[0] * Q[7]) * invd;
  iQ[8] = (Q[0] * Q[4] - Q[1] * Q[3]) * invd;
  float R[9];
  mm3(iQ, P, R);

  // ---- predicated squaring (branch-free, matches fori_loop unroll) ----
#pragma unroll
  for (int k = 0; k < 6; ++k) {
    float R2[9];
    mm3(R, R, R2);
    const bool sq = k < nsi;
#pragma unroll
    for (int i = 0; i < 9; ++i) R[i] = sq ? R2[i] : R[i];
  }

  // ---- quad = delta^T E delta ; det(A) on unscaled A ----
  float quad = 0.f;
#pragma unroll
  for (int r = 0; r < 3; ++r) {
    const float rowv = fmaf(R[3 * r + 0], dlt[0],
                       fmaf(R[3 * r + 1], dlt[1], R[3 * r + 2] * dlt[2]));
    quad = fmaf(dlt[r], rowv, quad);
  }
  const float detA = fmaf(m[0], m[4] * m[8] - m[5] * m[7],
                     fmaf(-m[1], m[3] * m[8] - m[5] * m[6],
                          m[2] * (m[3] * m[7] - m[4] * m[6])));

  float contrib = ok ? (quad + detA) : 0.0f;

  // ---- wave32 + cross-wave reduction -> one partial per block ----
#pragma unroll
  for (int off = 16; off >= 1; off >>= 1)
    contrib += __shfl_xor(contrib, off, 32);
  if ((t & 31) == 0) warpSums[t >> 5] = contrib;
  __syncthreads();
  if (t == 0) {
    float s = 0.f;
#pragma unroll
    for (int w = 0; w < TPB / 32; ++w) s += warpSums[w];
    blockSums[blockIdx.x] = s;
  }
}

// Final reduce: one wave32. Serial strided accumulate -> 32 lane sums,
// then V_WMMA_F32_16X16X4_F32 with an all-ones A folds lanes n and n+16
// (D[0][n] = s[n] + s[n+16]), finished with 16-lane xor-shuffles.
__global__ __launch_bounds__(32) void final_reduce_kernel(
    const float* __restrict__ blockSums, int nblk, float scale,
    float* __restrict__ out) {
  float s = 0.f;
  for (int i = threadIdx.x; i < nblk; i += 32) s += blockSums[i];

  v2f a; a.x = 1.0f; a.y = 1.0f;   // A = ones(16x4)
  v2f b; b.x = s;    b.y = 0.0f;   // B rows K=0,1 carry the 32 lane sums
  v8f c = {};
  c = __builtin_amdgcn_wmma_f32_16x16x4_f32(false, a, false, b, (short)0, c,
                                            false, false);
  float r = c[0];                  // lanes 0..15: D[0][n] = s[n] + s[n+16]
#pragma unroll
  for (int off = 8; off >= 1; off >>= 1) r += __shfl_xor(r, off, 32);
  if (threadIdx.x == 0) out[0] = r * scale;
}

extern "C" void kernel_launch(void* const* d_in, const int* in_sizes, int n_in,
                              void* d_out, int out_size, void* d_ws, size_t ws_size,
                              hipStream_t stream) {
  const float* mean = (const float*)d_in[0];
  const float* var  = (const float*)d_in[1];
  const float* gt   = (const float*)d_in[2];
  const long long total = (long long)in_sizes[1] / 9;   // bs*n matrices
  const int nblk = (int)((total + TPB - 1) / TPB);
  float* ws = (float*)d_ws;

  expm_loss_kernel<<<nblk, TPB, 0, stream>>>(mean, var, gt, ws, total);

  const float scale = 0.5f / (float)total;  // 0.5 * sum / (n*bs)
  final_reduce_kernel<<<1, 32, 0, stream>>>(ws, nblk, scale, (float*)d_out);
}